// SolveScheduling_62826781606434
// MI455X (gfx1250) — compile-verified
//
#include <hip/hip_runtime.h>
#include <hip/hip_bf16.h>

// SQP + primal-dual IPM scheduling solver for MI455X (gfx1250, wave32).
//
// Design:
//  * 1 wave = 16 scenarios. Lane (L & 15) owns scenario state in registers;
//    both lane halves carry bit-identical mirrored state (same loads, same
//    deterministic ops), so EXEC is all-1s for WMMA, the upper half feeds its
//    A-operand K-slice from its OWN registers (one v_cndmask per slice), and
//    duplicate LDS stores from the two halves are benign (same addr, same
//    value).
//  * KKT matrix G^T diag(w) G is EXACTLY tridiagonal (G = [D;-D], D = diff
//    operator) -> Thomas solve per lane instead of dense 24x24 linalg.solve.
//  * The four per-step G-GEMMs (z@G^T, lam@G, t@G, dz@G^T) run on the matrix
//    pipe via v_wmma_f32_16x16x4_f32. G entries are {0,+1,-1} and C is f32,
//    so WMMA results are bit-equivalent to the scalar stencil.
//  * 16x16 f32 D tiles (batch striped over VGPRs, N over lanes) are
//    transposed back to lane-major through LDS.
//  * 1/s is computed once per IPM step; dependent divisions become FMAs.

#define NVAR 24
#define MCON 46
#define TILE_B 16

typedef float v2f __attribute__((ext_vector_type(2)));
typedef float v8f __attribute__((ext_vector_type(8)));

// Ramp constraint matrix G[m][i], m in [0,46), i in [0,24).
// Rows 0..22:  +1 at i=m, -1 at i=m+1   (D)
// Rows 23..45: -1 at i=m-23, +1 at i=m-22  (-D)
// Out of range -> 0 (used for K/N padding).
__device__ __forceinline__ float Gmat(int m, int i) {
    if (m < 23)  return (i == m      ? 1.0f : 0.0f) - (i == m + 1  ? 1.0f : 0.0f);
    if (m < 46)  return (i == m - 22 ? 1.0f : 0.0f) - (i == m - 23 ? 1.0f : 0.0f);
    return 0.0f;
}

// result[b, m] = sum_i x[b, i] * G[m, i]   (x @ G^T), x length 24.
// D tiles written to ldsT[b*48 + m] for b in [0,16), m in [0,48).
// x[] is mirrored across lane halves -> upper half reads K+2 from own regs.
__device__ __forceinline__ void gemm_xGT(const float* x, float* ldsT, int lane) {
    const int  src = lane & 15;
    const bool lo  = lane < 16;
    v8f acc[3] = {};
#pragma unroll
    for (int ks = 0; ks < 6; ++ks) {           // K = 24 -> 6 slices of 4
        v2f a;
        a.x = lo ? x[4 * ks + 0] : x[4 * ks + 2];
        a.y = lo ? x[4 * ks + 1] : x[4 * ks + 3];
        const int kb = 4 * ks + (lo ? 0 : 2);  // this half-wave's K base
#pragma unroll
        for (int nt = 0; nt < 3; ++nt) {       // N = 46 -> 3 tiles of 16 (pad 48)
            const int n = nt * 16 + src;       // n == constraint row m
            v2f b;
            b.x = Gmat(n, kb);
            b.y = Gmat(n, kb + 1);
            acc[nt] = __builtin_amdgcn_wmma_f32_16x16x4_f32(
                false, a, false, b, (short)0, acc[nt], false, false);
        }
    }
#pragma unroll
    for (int nt = 0; nt < 3; ++nt)
#pragma unroll
        for (int r = 0; r < 8; ++r)
            ldsT[(r + (lo ? 0 : 8)) * 48 + nt * 16 + src] = acc[nt][r];
}

// result[b, i] = sum_m t[b, m] * G[m, i]   (t @ G), t length 48 (t[46..47]=0).
// D tiles written to ldsT[b*48 + i] for i in [0,32) (only i<24 meaningful).
__device__ __forceinline__ void gemm_tG(const float* t, float* ldsT, int lane) {
    const int  src = lane & 15;
    const bool lo  = lane < 16;
    v8f acc[2] = {};
#pragma unroll
    for (int ks = 0; ks < 12; ++ks) {          // K = 46 -> 12 slices of 4 (pad 48)
        v2f a;
        a.x = lo ? t[4 * ks + 0] : t[4 * ks + 2];
        a.y = lo ? t[4 * ks + 1] : t[4 * ks + 3];
        const int kb = 4 * ks + (lo ? 0 : 2);  // k == constraint row m
#pragma unroll
        for (int nt = 0; nt < 2; ++nt) {       // N = 24 -> 2 tiles of 16 (pad 32)
            const int n = nt * 16 + src;       // n == variable index i
            v2f b;
            b.x = Gmat(kb, n);
            b.y = Gmat(kb + 1, n);
            acc[nt] = __builtin_amdgcn_wmma_f32_16x16x4_f32(
                false, a, false, b, (short)0, acc[nt], false, false);
        }
    }
#pragma unroll
    for (int nt = 0; nt < 2; ++nt)
#pragma unroll
        for (int r = 0; r < 8; ++r)
            ldsT[(r + (lo ? 0 : 8)) * 48 + nt * 16 + src] = acc[nt][r];
}

__global__ __launch_bounds__(32)
void SolveScheduling_62826781606434_kernel(const float* __restrict__ mu_g,
                                           const float* __restrict__ sig_g,
                                           float* __restrict__ out) {
    __shared__ float ldsT[TILE_B * 48];   // WMMA D-tile transpose buffer
    __shared__ float ldsRP[TILE_B * 48];  // r_prim stash

    const int lane = threadIdx.x;          // 0..31
    const int src  = lane & 15;
    const int bat  = blockIdx.x * TILE_B + src;   // scenario (mirrored in halves)

    const float H_RAMP  = 0.4f;
    const float SIGMA_C = 0.1f;
    const float GSUM    = 50.5f;           // G_UNDER + G_OVER
    const float GUNDER  = 50.0f;
    const float INV_SQRT2   = 0.70710678118654752f;
    const float INV_SQRT2PI = 0.39894228040143268f;

    float z0[NVAR], zq[NVAR];
#pragma unroll
    for (int i = 0; i < NVAR; ++i) z0[i] = mu_g[bat * NVAR + i];  // SQP init

    for (int sqp = 0; sqp < 21; ++sqp) {   // 20 SQP iters + 1 final solve
        // Linearize penalty terms at z0: q = d2g + 1, p = dg - d2g*z0 - mu.
        // mu/sig are re-read each SQP iter (L2-resident) to keep VGPRs low.
        float q[NVAR], p[NVAR];
#pragma unroll
        for (int i = 0; i < NVAR; ++i) {
            float mu  = mu_g[bat * NVAR + i];
            float sg  = sig_g[bat * NVAR + i];
            float t   = (z0[i] - mu) / sg;
            float cdf = 0.5f * (1.0f + erff(t * INV_SQRT2));
            float pdf = expf(-0.5f * t * t) * (INV_SQRT2PI / sg);
            float dg  = GSUM * cdf - GUNDER;
            float d2g = GSUM * pdf;
            q[i] = d2g + 1.0f;
            p[i] = dg - d2g * z0[i] - mu;
        }

        // IPM state
        float s[MCON], lam[MCON];
#pragma unroll
        for (int m = 0; m < MCON; ++m) { s[m] = 1.0f; lam[m] = 1.0f; }
#pragma unroll
        for (int i = 0; i < NVAR; ++i) zq[i] = 0.0f;

        for (int it = 0; it < 25; ++it) {
            // ---- r_prim = z @ G^T + s - h  (WMMA) ----
            gemm_xGT(zq, ldsT, lane);
            __syncthreads();
            float musum = 0.0f;
            float rs[MCON];                      // 1/s, hoisted once per step
#pragma unroll
            for (int m = 0; m < MCON; ++m) {
                float rp = ldsT[src * 48 + m] + s[m] - H_RAMP;
                ldsRP[src * 48 + m] = rp;        // both halves: same addr+value
                musum += s[m] * lam[m];
                rs[m] = 1.0f / s[m];
            }
            const float sigmu = SIGMA_C * (musum * (1.0f / (float)MCON));
            __syncthreads();

            // ---- glam = lam @ G  (WMMA) -> r_dual ----
            float t48[48];
#pragma unroll
            for (int m = 0; m < MCON; ++m) t48[m] = lam[m];
            t48[46] = 0.0f; t48[47] = 0.0f;
            gemm_tG(t48, ldsT, lane);
            __syncthreads();
            float rhs[NVAR];
#pragma unroll
            for (int i = 0; i < NVAR; ++i)
                rhs[i] = q[i] * zq[i] + p[i] + ldsT[src * 48 + i];  // r_dual
            __syncthreads();

            // ---- gt = ((lam*r_prim - r_cent)/s) @ G  (WMMA) -> rhs ----
            // t = (lam*rp - (s*lam - sigmu))/s = (lam*(rp - s) + sigmu) * (1/s)
#pragma unroll
            for (int m = 0; m < MCON; ++m) {
                float rp = ldsRP[src * 48 + m];
                t48[m] = (lam[m] * (rp - s[m]) + sigmu) * rs[m];
            }
            gemm_tG(t48, ldsT, lane);
            __syncthreads();
#pragma unroll
            for (int i = 0; i < NVAR; ++i)
                rhs[i] = -(rhs[i] + ldsT[src * 48 + i]);
            __syncthreads();

            // ---- Tridiagonal KKT solve (Thomas). M = diag(q) + tridiag(v),
            //      v[k] = lam[k]/s[k] + lam[23+k]/s[23+k]; SPD & diag-dominant.
            float v[23];
#pragma unroll
            for (int k = 0; k < 23; ++k)
                v[k] = lam[k] * rs[k] + lam[23 + k] * rs[23 + k];
            float cp[23], dp[NVAR], dz[NVAR];
            {
                float r0 = 1.0f / (q[0] + v[0]);
                cp[0] = -v[0] * r0;
                dp[0] = rhs[0] * r0;
#pragma unroll
                for (int i = 1; i < NVAR; ++i) {
                    float offm  = -v[i - 1];
                    float diag  = q[i] + v[i - 1] + ((i < 23) ? v[i] : 0.0f);
                    float rden  = 1.0f / (diag - offm * cp[i - 1]);
                    if (i < 23) cp[i] = -v[i] * rden;
                    dp[i] = (rhs[i] - offm * dp[i - 1]) * rden;
                }
                dz[23] = dp[23];
#pragma unroll
                for (int i = 22; i >= 0; --i) dz[i] = dp[i] - cp[i] * dz[i + 1];
            }

            // ---- gdz = dz @ G^T  (WMMA) ----
            gemm_xGT(dz, ldsT, lane);
            __syncthreads();

            // ---- step length ----
            float amin = 1.0e9f;
#pragma unroll
            for (int m = 0; m < MCON; ++m) {
                float gd = ldsT[src * 48 + m];
                float rp = ldsRP[src * 48 + m];
                float ds = -(rp + gd);
                float rc = s[m] * lam[m] - sigmu;
                float dl = (-rc - lam[m] * ds) * rs[m];
                if (ds < 0.0f) amin = fminf(amin, -s[m] / ds);
                if (dl < 0.0f) amin = fminf(amin, -lam[m] / dl);
            }
            const float alpha = fminf(1.0f, 0.99f * amin);

            // ---- updates (recompute ds/dlam; uses pre-update s,lam) ----
#pragma unroll
            for (int m = 0; m < MCON; ++m) {
                float gd = ldsT[src * 48 + m];
                float rp = ldsRP[src * 48 + m];
                float ds = -(rp + gd);
                float rc = s[m] * lam[m] - sigmu;
                float dl = (-rc - lam[m] * ds) * rs[m];
                s[m]   += alpha * ds;
                lam[m] += alpha * dl;
            }
#pragma unroll
            for (int i = 0; i < NVAR; ++i) zq[i] += alpha * dz[i];
            __syncthreads();   // ldsT reused next iteration
        }

#pragma unroll
        for (int i = 0; i < NVAR; ++i) z0[i] = zq[i];
    }

    if (lane < 16) {
#pragma unroll
        for (int i = 0; i < NVAR; ++i) out[bat * NVAR + i] = zq[i];
    }
}

extern "C" void kernel_launch(void* const* d_in, const int* in_sizes, int n_in,
                              void* d_out, int out_size, void* d_ws, size_t ws_size,
                              hipStream_t stream) {
    const float* mu  = (const float*)d_in[0];
    const float* sig = (const float*)d_in[1];
    float* out = (float*)d_out;
    const int nb = in_sizes[0] / NVAR;       // 1024 scenarios
    const int blocks = nb / TILE_B;          // 16 scenarios per wave/block
    SolveScheduling_62826781606434_kernel<<<blocks, 32, 0, stream>>>(mu, sig, out);
}